// Transformer_87574383165740
// MI455X (gfx1250) — compile-verified
//
#include <hip/hip_runtime.h>
#include <math.h>

// Problem constants (from the reference)
#define B_  16
#define S_  100
#define D_  512
#define T_  60
#define V_  32000
#define R_  30
#define H_  8
#define HD_ 64
#define BT_ (B_*T_)

typedef float v2f __attribute__((ext_vector_type(2)));
typedef float v8f __attribute__((ext_vector_type(8)));

__device__ __forceinline__ float sigmoidf_(float x){ return 1.0f/(1.0f+expf(-x)); }

// ---------------------------------------------------------------------------
// 1) temp softmax table: temp[t][k], k=0..19.  L=min(t+1,20), slots k>=20-L.
// ---------------------------------------------------------------------------
__global__ void k_temp(float* __restrict__ temp_tab){
  int t = threadIdx.x;
  if (t >= T_) return;
  int L = (t+1 < 20) ? t+1 : 20;
  int kmin = 20 - L;
  float e[20]; float s = 0.f;
  for (int k = 0; k < 20; ++k){
    float w = (k >= kmin) ? expf((float)(k+1)/20.0f) : 0.0f;
    e[k] = w; s += w;
  }
  float inv = 1.0f/s;
  for (int k = 0; k < 20; ++k) temp_tab[t*20+k] = e[k]*inv;
}

// ---------------------------------------------------------------------------
// 2) attn_w[b,t,s] and query_s[b,t,d]: temp-weighted 20-step history windows.
//    buf slot k at step t holds row (t-19+k); zeros for negative indices.
// ---------------------------------------------------------------------------
__global__ void k_bufmix(const float* __restrict__ attn_seq,
                         const float* __restrict__ word_states,
                         const float* __restrict__ temp_tab,
                         float* __restrict__ attn_w,
                         float* __restrict__ query_s){
  int bt = blockIdx.x; int b = bt / T_, t = bt % T_;
  __shared__ float tw[20];
  if (threadIdx.x < 20) tw[threadIdx.x] = temp_tab[t*20 + threadIdx.x];
  __syncthreads();
  int kmin = (t < 19) ? (19 - t) : 0;
  for (int s = threadIdx.x; s < S_; s += blockDim.x){
    float acc = 0.f;
    for (int k = kmin; k < 20; ++k)
      acc += tw[k]*attn_seq[((size_t)b*T_ + (t-19+k))*S_ + s];
    attn_w[(size_t)bt*S_ + s] = acc;
  }
  for (int d = threadIdx.x; d < D_; d += blockDim.x){
    float acc = 0.f;
    for (int k = kmin; k < 20; ++k)
      acc += tw[k]*word_states[((size_t)b*T_ + (t-19+k))*D_ + d];
    query_s[(size_t)bt*D_ + d] = acc;
  }
}

// ---------------------------------------------------------------------------
// 3) gate dots per (b,t): remove/add/ptr sigmoids on word·W + bias
// ---------------------------------------------------------------------------
__global__ void k_worddots(const float* __restrict__ word_states,
                           const float* __restrict__ W_remove, const float* __restrict__ b_remove,
                           const float* __restrict__ W_add,    const float* __restrict__ b_add,
                           const float* __restrict__ W_ptr,    const float* __restrict__ b_ptr,
                           float* __restrict__ remove_p, float* __restrict__ add_p,
                           float* __restrict__ p_gen){
  int bt = blockIdx.x;
  const float* w = word_states + (size_t)bt*D_;
  __shared__ float red[3][256];
  float s0=0.f, s1=0.f, s2=0.f;
  for (int d = threadIdx.x; d < D_; d += 256){
    float x = w[d];
    s0 += x*W_remove[d]; s1 += x*W_add[d]; s2 += x*W_ptr[d];
  }
  red[0][threadIdx.x]=s0; red[1][threadIdx.x]=s1; red[2][threadIdx.x]=s2;
  __syncthreads();
  for (int off=128; off>0; off>>=1){
    if (threadIdx.x < off){
      red[0][threadIdx.x]+=red[0][threadIdx.x+off];
      red[1][threadIdx.x]+=red[1][threadIdx.x+off];
      red[2][threadIdx.x]+=red[2][threadIdx.x+off];
    }
    __syncthreads();
  }
  if (threadIdx.x==0){
    remove_p[bt] = sigmoidf_(red[0][0]+b_remove[0]);
    add_p[bt]    = sigmoidf_(red[1][0]+b_add[0]);
    p_gen[bt]    = sigmoidf_(red[2][0]+b_ptr[0]);
  }
}

// ---------------------------------------------------------------------------
// 4) e1[b,s]=e·W_next[:D], e2[b,s]=e·W_next[D:]  (cat-GEMV split, step-invariant)
// ---------------------------------------------------------------------------
__global__ void k_eproj(const float* __restrict__ e_outputs,
                        const float* __restrict__ W_next,
                        float* __restrict__ e1, float* __restrict__ e2){
  int bs = blockIdx.x;
  const float* e = e_outputs + (size_t)bs*D_;
  __shared__ float r0[128], r1[128];
  float s0=0.f, s1=0.f;
  for (int d = threadIdx.x; d < D_; d += 128){
    float x = e[d];
    s0 += x*W_next[d]; s1 += x*W_next[D_+d];
  }
  r0[threadIdx.x]=s0; r1[threadIdx.x]=s1;
  __syncthreads();
  for (int off=64; off>0; off>>=1){
    if (threadIdx.x<off){ r0[threadIdx.x]+=r0[threadIdx.x+off]; r1[threadIdx.x]+=r1[threadIdx.x+off]; }
    __syncthreads();
  }
  if (threadIdx.x==0){ e1[bs]=r0[0]; e2[bs]=r1[0]; }
}

// ---------------------------------------------------------------------------
// 5) nxt[b,t,s] = sigmoid(e1[b,s] + (attn_w[b,t,:]·e2[b,:]) + b_next)
// ---------------------------------------------------------------------------
__global__ void k_nxt(const float* __restrict__ attn_w, const float* __restrict__ e1,
                      const float* __restrict__ e2, const float* __restrict__ b_next,
                      float* __restrict__ nxt){
  int bt = blockIdx.x; int b = bt / T_;
  __shared__ float red[128];
  float s = 0.f;
  for (int ss = threadIdx.x; ss < S_; ss += 128)
    s += attn_w[(size_t)bt*S_+ss]*e2[b*S_+ss];
  red[threadIdx.x]=s; __syncthreads();
  for (int off=64; off>0; off>>=1){ if (threadIdx.x<off) red[threadIdx.x]+=red[threadIdx.x+off]; __syncthreads(); }
  float lcw = red[0] + b_next[0];
  for (int ss = threadIdx.x; ss < S_; ss += 128)
    nxt[(size_t)bt*S_+ss] = sigmoidf_(e1[b*S_+ss] + lcw);
}

// ---------------------------------------------------------------------------
// 6) add_state scan (elementwise over [B,S], sequential in t — tiny):
//    gain_t = (1-a)*add_p_t*nxt_t ; a += gain_t ; a0 = add0[s]
// ---------------------------------------------------------------------------
__global__ void k_addscan(const float* __restrict__ add_p, const float* __restrict__ nxt,
                          float* __restrict__ gain){
  int idx = blockIdx.x*blockDim.x + threadIdx.x;
  if (idx >= B_*S_) return;
  int b = idx / S_, s = idx % S_;
  float a = (s < 50) ? (1.0f - (float)(s+1)/50.0f) : 0.0f;
  for (int t = 0; t < T_; ++t){
    float g = (1.0f - a)*add_p[b*T_+t]*nxt[((size_t)b*T_+t)*S_+s];
    gain[((size_t)b*T_+t)*S_+s] = g;
    a += g;
  }
}

// ---------------------------------------------------------------------------
// 7) fp32 WMMA GEMMs (V_WMMA_F32_16X16X4_F32, exact fp32 math).
//    One 16x16 output tile per wave, 8 waves per block.
//    A layout (32-bit 16x4): lane&15 = M row; half-wave owns K pair {0,1}/{2,3}.
//    C/D layout: VGPR j <-> row j + 8*(lane>=16), col = lane&15.
// ---------------------------------------------------------------------------
#define GEMM_WAVES 8

__global__ void __launch_bounds__(256) k_gemm_nn(
    const float* __restrict__ A, const float* __restrict__ Bm,
    float* __restrict__ Cm, int M, int N, int K){
  int lane = threadIdx.x & 31, wid = threadIdx.x >> 5;
  int tile = blockIdx.x*GEMM_WAVES + wid;
  int tilesN = N >> 4;
  if (tile >= (M >> 4)*tilesN) return;          // wave-uniform: EXEC all-ones for WMMA
  int tm = (tile / tilesN) << 4, tn = (tile % tilesN) << 4;
  int mr = lane & 15;
  int kh = (lane >> 4) << 1;                    // 0 or 2
  const v2f* pa = (const v2f*)(A + (size_t)(tm+mr)*K);
  v8f c = {};
  for (int k0 = 0; k0 < K; k0 += 4){
    v2f a = pa[(k0 + kh) >> 1];
    v2f b;
    b.x = Bm[(size_t)(k0+kh  )*N + tn + mr];
    b.y = Bm[(size_t)(k0+kh+1)*N + tn + mr];
    c = __builtin_amdgcn_wmma_f32_16x16x4_f32(false, a, false, b, (short)0, c, false, false);
  }
  int rbase = tm + ((lane >> 4) << 3);
  #pragma unroll
  for (int j = 0; j < 8; ++j)
    Cm[(size_t)(rbase + j)*N + tn + mr] = c[j];
}

// NT variant (B supplied row-major [N,K], contraction over contiguous K) + bias.
__global__ void __launch_bounds__(256) k_gemm_nt_bias(
    const float* __restrict__ A, const float* __restrict__ Bt,
    const float* __restrict__ bias, float* __restrict__ Cm,
    int M, int N, int K){
  int lane = threadIdx.x & 31, wid = threadIdx.x >> 5;
  int tile = blockIdx.x*GEMM_WAVES + wid;
  int tilesN = N >> 4;
  if (tile >= (M >> 4)*tilesN) return;
  int tm = (tile / tilesN) << 4, tn = (tile % tilesN) << 4;
  int mr = lane & 15;
  int kh = (lane >> 4) << 1;
  const v2f* pa = (const v2f*)(A  + (size_t)(tm+mr)*K);
  const v2f* pb = (const v2f*)(Bt + (size_t)(tn+mr)*K);
  v8f c = {};
  for (int k0 = 0; k0 < K; k0 += 4){
    __builtin_prefetch(pb + ((k0 + kh) >> 1) + 32, 0, 1);   // stream weight rows
    v2f a = pa[(k0 + kh) >> 1];
    v2f b = pb[(k0 + kh) >> 1];
    c = __builtin_amdgcn_wmma_f32_16x16x4_f32(false, a, false, b, (short)0, c, false, false);
  }
  float bv = bias[tn + mr];
  int rbase = tm + ((lane >> 4) << 3);
  #pragma unroll
  for (int j = 0; j < 8; ++j)
    Cm[(size_t)(rbase + j)*N + tn + mr] = c[j] + bv;
}

// ---------------------------------------------------------------------------
// 8) Memory-bank recurrence. One workgroup per batch; mem[b] (100x512 f32 =
//    205 KB) lives in CDNA5's 320 KB LDS for all 60 steps.
//    Per step: selected = attn_t·mem ; sim = sigma(mem·qW) ;
//              mem = mem*(1 - r*attn_w*sim) + e*gain.
// ---------------------------------------------------------------------------
__global__ void __launch_bounds__(512) k_memscan(
    const float* __restrict__ e_outputs, const float* __restrict__ attn_seq,
    const float* __restrict__ attn_w,    const float* __restrict__ qW,
    const float* __restrict__ remove_p,  const float* __restrict__ gain,
    float* __restrict__ selected){
  extern __shared__ float sm[];
  float* mem_s = sm;                 // S_*D_
  float* qw_s  = sm + S_*D_;         // D_
  float* ar_s  = qw_s + D_;          // S_  attn_t row
  float* aw_s  = ar_s + S_;          // S_  attn_w row
  float* g_s   = aw_s + S_;          // S_  gain row
  float* sim_s = g_s  + S_;          // S_
  int b = blockIdx.x, tid = threadIdx.x;

  for (int i = tid; i < S_*D_; i += 512){
    int s = i / D_;
    float a0 = (s < 50) ? (1.0f - (float)(s+1)/50.0f) : 0.0f;
    mem_s[i] = e_outputs[(size_t)b*S_*D_ + i]*a0;
  }
  __syncthreads();

  for (int t = 0; t < T_; ++t){
    int bt = b*T_ + t;
    if (tid < S_){
      ar_s[tid] = attn_seq[(size_t)bt*S_ + tid];
      aw_s[tid] = attn_w [(size_t)bt*S_ + tid];
      g_s [tid] = gain   [(size_t)bt*S_ + tid];
    }
    qw_s[tid] = qW[(size_t)bt*D_ + tid];          // blockDim==D_
    __syncthreads();

    // selected (pre-update mem): one column per thread
    {
      float acc = 0.f;
      for (int s = 0; s < S_; ++s) acc += ar_s[s]*mem_s[s*D_ + tid];
      selected[(size_t)bt*D_ + tid] = acc;
    }
    // sim per memory row: 16 waves x 32-lane dot reductions
    {
      int lane = tid & 31, wid = tid >> 5;
      for (int s = wid; s < S_; s += 16){
        float p = 0.f;
        for (int d = lane; d < D_; d += 32) p += mem_s[s*D_ + d]*qw_s[d];
        p += __shfl_down(p, 16);
        p += __shfl_down(p, 8);
        p += __shfl_down(p, 4);
        p += __shfl_down(p, 2);
        p += __shfl_down(p, 1);
        if (lane == 0) sim_s[s] = sigmoidf_(p);
      }
    }
    __syncthreads();

    // remove + add, fused
    float rp = remove_p[bt];
    for (int s = 0; s < S_; ++s){
      float scale = 1.0f - rp*aw_s[s]*sim_s[s];
      mem_s[s*D_ + tid] = mem_s[s*D_ + tid]*scale
                        + e_outputs[((size_t)b*S_ + s)*D_ + tid]*g_s[s];
    }
    __syncthreads();
  }
}

// ---------------------------------------------------------------------------
// 9) role attention: scores[h,r] = (q_h · k_{r,h})/8, mask, softmax_r, mean_h
// ---------------------------------------------------------------------------
__global__ void k_rolesel(const float* __restrict__ q_role, const float* __restrict__ k_role,
                          const int* __restrict__ rolename_mask,
                          float* __restrict__ role_attn){
  int bt = blockIdx.x; int b = bt / T_;
  __shared__ float sc[H_][R_];
  int tid = threadIdx.x;
  if (tid < H_*R_){
    int h = tid / R_, r = tid % R_;
    const float* q = q_role + (size_t)bt*D_ + h*HD_;
    const float* k = k_role + (size_t)(b*R_ + r)*D_ + h*HD_;
    float acc = 0.f;
    for (int d = 0; d < HD_; ++d) acc += q[d]*k[d];
    acc *= 0.125f;                                   // 1/sqrt(64)
    if (rolename_mask[b*R_ + r] == 0) acc = -1e9f;
    sc[h][r] = acc;
  }
  __syncthreads();
  if (tid < H_){
    float mx = -1e30f;
    for (int r = 0; r < R_; ++r) mx = fmaxf(mx, sc[tid][r]);
    float s = 0.f;
    for (int r = 0; r < R_; ++r){ float e = expf(sc[tid][r]-mx); sc[tid][r]=e; s+=e; }
    float inv = 1.0f/s;
    for (int r = 0; r < R_; ++r) sc[tid][r] *= inv;
  }
  __syncthreads();
  if (tid < R_){
    float s = 0.f;
    for (int h = 0; h < H_; ++h) s += sc[h][tid];
    role_attn[(size_t)bt*R_ + tid] = s*(1.0f/H_);
  }
}

// ---------------------------------------------------------------------------
// 10) in-place row softmax over V=32000, scaled by (1-p_gen)
// ---------------------------------------------------------------------------
__global__ void k_softmax_row(float* __restrict__ out, const float* __restrict__ p_gen){
  int bt = blockIdx.x;
  float* row = out + (size_t)bt*V_;
  __shared__ float red[256];
  float mx = -1e30f;
  for (int v = threadIdx.x; v < V_; v += 256) mx = fmaxf(mx, row[v]);
  red[threadIdx.x]=mx; __syncthreads();
  for (int off=128; off>0; off>>=1){ if (threadIdx.x<off) red[threadIdx.x]=fmaxf(red[threadIdx.x],red[threadIdx.x+off]); __syncthreads(); }
  mx = red[0]; __syncthreads();
  float s = 0.f;
  for (int v = threadIdx.x; v < V_; v += 256) s += expf(row[v]-mx);
  red[threadIdx.x]=s; __syncthreads();
  for (int off=128; off>0; off>>=1){ if (threadIdx.x<off) red[threadIdx.x]+=red[threadIdx.x+off]; __syncthreads(); }
  float scale = (1.0f - p_gen[bt])/red[0];
  for (int v = threadIdx.x; v < V_; v += 256) row[v] = expf(row[v]-mx)*scale;
}

// ---------------------------------------------------------------------------
// 11) pointer scatter; reference uses .at[].set -> last duplicate token wins
// ---------------------------------------------------------------------------
__global__ void k_scatter(float* __restrict__ out, const int* __restrict__ rolename,
                          const float* __restrict__ role_attn, const float* __restrict__ p_gen){
  int bt = blockIdx.x; int b = bt / T_;
  int r = threadIdx.x;
  if (r >= R_) return;
  int tok = rolename[b*R_ + r];
  for (int r2 = r+1; r2 < R_; ++r2)
    if (rolename[b*R_ + r2] == tok) return;          // a later write would override
  out[(size_t)bt*V_ + tok] += role_attn[(size_t)bt*R_ + r]*p_gen[bt];
}

// ---------------------------------------------------------------------------
extern "C" void kernel_launch(void* const* d_in, const int* in_sizes, int n_in,
                              void* d_out, int out_size, void* d_ws, size_t ws_size,
                              hipStream_t stream){
  (void)in_sizes; (void)n_in; (void)out_size; (void)ws_size;
  const float* e_outputs   = (const float*)d_in[0];
  const float* r_outputs   = (const float*)d_in[1];
  const float* word_states = (const float*)d_in[2];
  const float* attn_seq    = (const float*)d_in[3];
  const float* W_remove    = (const float*)d_in[4];
  const float* b_remove    = (const float*)d_in[5];
  const float* W_add       = (const float*)d_in[6];
  const float* b_add       = (const float*)d_in[7];
  const float* W_q         = (const float*)d_in[8];
  const float* W_next      = (const float*)d_in[9];
  const float* b_next      = (const float*)d_in[10];
  const float* W_ptr       = (const float*)d_in[11];
  const float* b_ptr       = (const float*)d_in[12];
  const float* logit_W     = (const float*)d_in[13];
  const float* logit_b     = (const float*)d_in[14];
  const float* Wq_role     = (const float*)d_in[15];
  const float* Wk_role     = (const float*)d_in[16];
  const int*   rolename    = (const int*)d_in[17];
  const int*   rolename_mask = (const int*)d_in[18];
  float* out = (float*)d_out;

  // workspace carve-out (~10.1 MB of floats)
  float* ws = (float*)d_ws;
  float* temp_tab = ws; ws += T_*20;
  float* attn_w   = ws; ws += (size_t)BT_*S_;
  float* query_s  = ws; ws += (size_t)BT_*D_;
  float* qW       = ws; ws += (size_t)BT_*D_;
  float* kro      = ws; ws += (size_t)B_*R_*D_;
  float* qro      = ws; ws += (size_t)BT_*D_;
  float* sel      = ws; ws += (size_t)BT_*D_;
  float* remove_p = ws; ws += BT_;
  float* add_p    = ws; ws += BT_;
  float* p_gen    = ws; ws += BT_;
  float* e1       = ws; ws += B_*S_;
  float* e2       = ws; ws += B_*S_;
  float* nxt      = ws; ws += (size_t)BT_*S_;
  float* gain     = ws; ws += (size_t)BT_*S_;
  float* role_attn= ws; ws += (size_t)BT_*R_;

  k_temp   <<<1, 64, 0, stream>>>(temp_tab);
  k_bufmix <<<BT_, 256, 0, stream>>>(attn_seq, word_states, temp_tab, attn_w, query_s);
  k_worddots<<<BT_, 256, 0, stream>>>(word_states, W_remove, b_remove, W_add, b_add,
                                      W_ptr, b_ptr, remove_p, add_p, p_gen);
  k_eproj  <<<B_*S_, 128, 0, stream>>>(e_outputs, W_next, e1, e2);
  k_nxt    <<<BT_, 128, 0, stream>>>(attn_w, e1, e2, b_next, nxt);
  k_addscan<<<(B_*S_+255)/256, 256, 0, stream>>>(add_p, nxt, gain);

  { // qW = query_s @ W_q   [960x512]x[512x512]
    int tiles = (BT_/16)*(D_/16);
    k_gemm_nn<<<(tiles+GEMM_WAVES-1)/GEMM_WAVES, 256, 0, stream>>>(query_s, W_q, qW, BT_, D_, D_);
  }
  { // k_role = r_outputs @ Wk_role   [480x512]x[512x512]
    int tiles = (B_*R_/16)*(D_/16);
    k_gemm_nn<<<(tiles+GEMM_WAVES-1)/GEMM_WAVES, 256, 0, stream>>>(r_outputs, Wk_role, kro, B_*R_, D_, D_);
  }

  size_t lds_bytes = (size_t)(S_*D_ + D_ + 4*S_)*sizeof(float);   // ~208 KB < 320 KB
  k_memscan<<<B_, 512, lds_bytes, stream>>>(e_outputs, attn_seq, attn_w, qW,
                                            remove_p, gain, sel);

  { // q_role = selected @ Wq_role
    int tiles = (BT_/16)*(D_/16);
    k_gemm_nn<<<(tiles+GEMM_WAVES-1)/GEMM_WAVES, 256, 0, stream>>>(sel, Wq_role, qro, BT_, D_, D_);
  }
  k_rolesel<<<BT_, 256, 0, stream>>>(qro, kro, rolename_mask, role_attn);

  { // logits = word_states @ logit_W^T + logit_b   [960x512]x[32000x512]^T
    int tiles = (BT_/16)*(V_/16);
    k_gemm_nt_bias<<<(tiles+GEMM_WAVES-1)/GEMM_WAVES, 256, 0, stream>>>(
        word_states, logit_W, logit_b, out, BT_, V_, D_);
  }
  k_softmax_row<<<BT_, 256, 0, stream>>>(out, p_gen);
  k_scatter    <<<BT_, 32, 0, stream>>>(out, rolename, role_attn, p_gen);
}